// DCNN_73993696576081
// MI455X (gfx1250) — compile-verified
//
#include <hip/hip_runtime.h>
#include <hip/hip_bf16.h>
#include <math.h>

typedef __attribute__((ext_vector_type(2))) float v2f;
typedef __attribute__((ext_vector_type(8))) float v8f;

#define S_LEN 2048
#define L1    2054          // conv1 output length = S + 6
#define EMBD  50
#define C1    6
#define KW1   7
#define KPAD  64            // channels padded 50 -> 64 so tap=k>>6, ch=k&63
#define NKP   (KW1 * KPAD)  // 448 padded K
#define NPAD  16            // output channels padded 6 -> 16 (full WMMA N tile)
#define NT    129           // ceil(L1 / 16) position tiles
#define BATCH 256

// ---------------------------------------------------------------------------
// Kernel 1: fused embedding-gather + conv1 (implicit GEMM on V_WMMA_F32_16X16X4_F32)
// One wave (32 lanes) computes a 16-position x 6-channel output tile.
// A (16x4 f32): lane m=lane&15, hi=lane>>4; VGPR j holds K = k0 + 2*hi + j.
// B (4x16 f32): mirrored mapping; lane n=lane&15 holds column n.
// C/D (16x16 f32): lanes 0-15 -> M=i, lanes 16-31 -> M=8+i, N=lane&15.
// W1 is staged into LDS padded to 16 columns so the B-fragment load is an
// unconditional ds_load_b32 (no per-lane EXEC predication in the hot loop).
// ---------------------------------------------------------------------------
__global__ __launch_bounds__(128)
void dcnn_conv1_wmma(const int* __restrict__ x,
                     const float* __restrict__ emb,
                     const float* __restrict__ W1,
                     const float* __restrict__ b1,
                     float* __restrict__ sig)
{
    __shared__ float w1s[NKP * NPAD];     // [tap*64+ch][n], n>=6 or ch>=50 -> 0
    __shared__ float embs[4][22 * KPAD];  // per-wave 22 token rows, zero padded

    const int tid     = threadIdx.x;
    const int lane    = tid & 31;
    const int wave    = tid >> 5;
    const int b       = blockIdx.y;
    const int tileIdx = blockIdx.x * 4 + wave;
    const int t0      = tileIdx * 16;

    // Cooperative load of zero-padded W1 into LDS (shared by all 4 waves).
    for (int idx = tid; idx < NKP * NPAD; idx += 128) {
        const int kk  = idx >> 4;
        const int n   = idx & 15;
        const int tap = kk >> 6, ch = kk & 63;
        float v = 0.f;
        if (n < C1 && ch < EMBD) v = W1[(tap * EMBD + ch) * C1 + n];
        w1s[idx] = v;
    }

    // Per-wave staging of the 22 embedding rows this tile touches
    // (rows p = t0-6 .. t0+15; OOB rows and padded channels are zero).
    float* embW = embs[wave];
    if (tileIdx < NT) {
        for (int idx = lane; idx < 22 * KPAD; idx += 32) {
            const int r = idx >> 6, c = idx & 63;
            const int p = t0 - 6 + r;
            float v = 0.f;
            if (c < EMBD && p >= 0 && p < S_LEN) {
                const int tok = x[b * S_LEN + p];
                v = emb[tok * EMBD + c];
            }
            embW[idx] = v;
        }
    }
    __syncthreads();

    if (tileIdx >= NT) return;   // whole wave exits together; EXEC stays full for WMMA

    const int m  = lane & 15;    // output position within tile (A row)
    const int hi = lane >> 4;
    const int n  = lane & 15;    // output channel (B column), valid n<6

    const float* aBase = &embW[(m << 6)];
    const float* bBase = &w1s[n];

    v8f acc = {};
    #pragma unroll 4
    for (int k0 = 0; k0 < NKP; k0 += 4) {
        const int kk0 = k0 + hi * 2;
        // patch(m, kk) = embW[(m + (kk>>6))*64 + (kk&63)] = embW[m*64 + kk]
        v2f a, bm;
        a[0]  = aBase[kk0];
        a[1]  = aBase[kk0 + 1];
        bm[0] = bBase[(kk0)     << 4];   // w1s[kk0*16 + n]
        bm[1] = bBase[(kk0 + 1) << 4];   // w1s[(kk0+1)*16 + n]
        acc = __builtin_amdgcn_wmma_f32_16x16x4_f32(
            /*neg_a=*/false, a, /*neg_b=*/false, bm,
            /*c_mod=*/(short)0, acc, /*reuse_a=*/false, /*reuse_b=*/false);
    }

    const float bias = (n < C1) ? b1[n] : 0.f;
    #pragma unroll
    for (int i = 0; i < 8; ++i) {
        const int t = t0 + (hi ? 8 : 0) + i;
        if (n < C1 && t < L1)
            sig[((size_t)b * L1 + t) * C1 + n] = acc[i] + bias;
    }
}

// ---------------------------------------------------------------------------
// lax.top_k ordering: descending value, ties broken by lower index.
// ---------------------------------------------------------------------------
__device__ __forceinline__ bool beats(float av, int ai, float bv, int bi) {
    return (av > bv) || (av == bv && ai < bi);
}

__device__ __forceinline__ float sigmoidf(float v) {
    return 1.0f / (1.0f + expf(-v));
}

// ---------------------------------------------------------------------------
// Kernel 2: per-batch top-8 pool -> sigmoid -> conv2 -> top-4 pool -> dense
// One 256-thread block per batch row.
// ---------------------------------------------------------------------------
__global__ __launch_bounds__(256)
void dcnn_tail(const float* __restrict__ sig,
               const float* __restrict__ W2,
               const float* __restrict__ b2,
               const float* __restrict__ Wd,
               const float* __restrict__ bd,
               float* __restrict__ dense)
{
    __shared__ float lv[256 * 8];
    __shared__ int   li[256 * 8];
    const int tid = threadIdx.x;
    const int b   = blockIdx.x;

    // Per-thread sorted top-8 over strided positions.
    float vals[8]; int idxs[8];
    #pragma unroll
    for (int j = 0; j < 8; ++j) { vals[j] = -__builtin_inff(); idxs[j] = 0x7fffffff; }

    for (int t = tid; t < L1; t += 256) {
        const float* p = &sig[((size_t)b * L1 + t) * C1];
        float s = 0.f;
        #pragma unroll
        for (int c = 0; c < C1; ++c) s += p[c];
        const float e = s * (1.0f / C1);
        if (beats(e, t, vals[7], idxs[7])) {
            int j = 7;
            while (j > 0 && beats(e, t, vals[j - 1], idxs[j - 1])) {
                vals[j] = vals[j - 1]; idxs[j] = idxs[j - 1]; --j;
            }
            vals[j] = e; idxs[j] = t;
        }
    }
    #pragma unroll
    for (int j = 0; j < 8; ++j) { lv[tid * 8 + j] = vals[j]; li[tid * 8 + j] = idxs[j]; }
    __syncthreads();

    // Tree-merge sorted 8-lists (indices are disjoint across threads).
    for (int s = 128; s > 0; s >>= 1) {
        if (tid < s) {
            float ov[8]; int oi[8];
            #pragma unroll
            for (int j = 0; j < 8; ++j) { ov[j] = lv[(tid + s) * 8 + j]; oi[j] = li[(tid + s) * 8 + j]; }
            float mv[8]; int mi[8];
            int pa = 0, pb = 0;
            #pragma unroll
            for (int k = 0; k < 8; ++k) {
                if (beats(vals[pa], idxs[pa], ov[pb], oi[pb])) { mv[k] = vals[pa]; mi[k] = idxs[pa]; ++pa; }
                else                                           { mv[k] = ov[pb];   mi[k] = oi[pb];   ++pb; }
            }
            #pragma unroll
            for (int j = 0; j < 8; ++j) { vals[j] = mv[j]; idxs[j] = mi[j]; }
        }
        __syncthreads();
        if (tid < s) {
            #pragma unroll
            for (int j = 0; j < 8; ++j) { lv[tid * 8 + j] = vals[j]; li[tid * 8 + j] = idxs[j]; }
        }
        __syncthreads();
    }

    // Tiny tail on thread 0 (~5 KFLOP).
    if (tid == 0) {
        float s1p[16][C1];
        for (int r = 0; r < 16; ++r)
            for (int c = 0; c < C1; ++c) s1p[r][c] = 0.f;
        for (int j = 0; j < 8; ++j) {
            const int t = idxs[j];   // gathered in top_k (value-descending) order
            for (int c = 0; c < C1; ++c)
                s1p[4 + j][c] = sigmoidf(sig[((size_t)b * L1 + t) * C1 + c]);
        }

        float out2[12][14];
        float ex2[12];
        for (int t = 0; t < 12; ++t) {
            float es = 0.f;
            for (int co = 0; co < 14; ++co) {
                float acc = b2[co];
                for (int tap = 0; tap < 5; ++tap)
                    for (int ci = 0; ci < C1; ++ci)
                        acc += s1p[t + tap][ci] * W2[(tap * C1 + ci) * 14 + co];
                out2[t][co] = acc;
                es += acc;
            }
            ex2[t] = es * (1.0f / 14.0f);
        }

        float tv[4]; int ti[4];
        for (int j = 0; j < 4; ++j) { tv[j] = -__builtin_inff(); ti[j] = 0x7fffffff; }
        for (int t = 0; t < 12; ++t) {
            if (beats(ex2[t], t, tv[3], ti[3])) {
                int j = 3;
                while (j > 0 && beats(ex2[t], t, tv[j - 1], ti[j - 1])) {
                    tv[j] = tv[j - 1]; ti[j] = ti[j - 1]; --j;
                }
                tv[j] = ex2[t]; ti[j] = t;
            }
        }

        float d = bd[0];
        for (int co = 0; co < 14; ++co) {
            float f = 0.f;
            for (int j = 0; j < 4; ++j) f += out2[ti[j]][co];
            d += (f * 0.25f) * Wd[co];
        }
        dense[b] = d;
    }
}

// ---------------------------------------------------------------------------
// Kernel 3: mean over batch + final sigmoid -> scalar output
// ---------------------------------------------------------------------------
__global__ __launch_bounds__(256)
void dcnn_final(const float* __restrict__ dense, float* __restrict__ out)
{
    __shared__ float red[256];
    const int tid = threadIdx.x;
    red[tid] = dense[tid];
    __syncthreads();
    for (int s = 128; s > 0; s >>= 1) {
        if (tid < s) red[tid] += red[tid + s];
        __syncthreads();
    }
    if (tid == 0) out[0] = sigmoidf(red[0] * (1.0f / BATCH));
}

// ---------------------------------------------------------------------------
extern "C" void kernel_launch(void* const* d_in, const int* in_sizes, int n_in,
                              void* d_out, int out_size, void* d_ws, size_t ws_size,
                              hipStream_t stream)
{
    (void)in_sizes; (void)n_in; (void)out_size; (void)ws_size;

    const int*   x   = (const int*)  d_in[0];
    const float* emb = (const float*)d_in[1];
    const float* W1  = (const float*)d_in[2];
    const float* b1  = (const float*)d_in[3];
    const float* W2  = (const float*)d_in[4];
    const float* b2  = (const float*)d_in[5];
    const float* Wd  = (const float*)d_in[6];
    const float* bd  = (const float*)d_in[7];

    float* sig   = (float*)d_ws;                       // [256][2054][6] f32 (~12.6 MB)
    float* dense = sig + (size_t)BATCH * L1 * C1;      // [256] f32

    dim3 g1((NT + 3) / 4, BATCH);
    dcnn_conv1_wmma<<<g1, 128, 0, stream>>>(x, emb, W1, b1, sig);
    dcnn_tail<<<BATCH, 256, 0, stream>>>(sig, W2, b2, Wd, bd, dense);
    dcnn_final<<<1, 256, 0, stream>>>(dense, (float*)d_out);
}